// PositionalEncoding_9328668967780
// MI455X (gfx1250) — compile-verified
//
#include <hip/hip_runtime.h>

// Problem constants (from reference)
#define DIM   256
#define SEQ   4096
#define BATCH 32

typedef float v4f __attribute__((ext_vector_type(4)));

// ---------------------------------------------------------------------------
// Kernel A: per-row prefix sum of valid(=!mask) -> pos[b,s]
//   pos = valid ? cumsum(valid)-1 : 0   (general; does not assume tail-pad)
// One 1024-thread block per batch row; each thread scans 4 mask bytes.
// ---------------------------------------------------------------------------
__global__ __launch_bounds__(1024)
void pe_pos_scan_kernel(const unsigned char* __restrict__ mask,
                        int* __restrict__ pos) {
  __shared__ int sums[1024];

  const int b = blockIdx.x;         // batch row
  const int t = threadIdx.x;        // 0..1023, covers 4096 = 1024*4 slots

  const unsigned char* mrow = mask + (size_t)b * SEQ;
  int* prow = pos + (size_t)b * SEQ;

  // 4 consecutive mask bytes per thread (rows are 4096B, so 4B-aligned).
  uchar4 m4 = *reinterpret_cast<const uchar4*>(mrow + t * 4);
  const int v0 = m4.x ? 0 : 1;
  const int v1 = m4.y ? 0 : 1;
  const int v2 = m4.z ? 0 : 1;
  const int v3 = m4.w ? 0 : 1;
  const int local = v0 + v1 + v2 + v3;

  sums[t] = local;
  __syncthreads();

  // Hillis-Steele inclusive scan over the 1024 per-thread sums.
  #pragma unroll
  for (int off = 1; off < 1024; off <<= 1) {
    int add = (t >= off) ? sums[t - off] : 0;
    __syncthreads();
    sums[t] += add;
    __syncthreads();
  }

  // Exclusive prefix before this thread's 4-element chunk.
  int c = sums[t] - local;

  int4 p;
  c += v0; p.x = v0 ? (c - 1) : 0;
  c += v1; p.y = v1 ? (c - 1) : 0;
  c += v2; p.z = v2 ? (c - 1) : 0;
  c += v3; p.w = v3 ? (c - 1) : 0;
  *reinterpret_cast<int4*>(prow + t * 4) = p;
}

// ---------------------------------------------------------------------------
// Kernel B: out[b,s,:] = x[b,s,:] + pe[pos[b,s],:]
// 256 threads/block = 8 wave32s; 64 lanes per token (64 * float4 = 256 dims),
// 4 tokens per block. x/out streamed with non-temporal hints so the 4 MiB pe
// table stays resident in the 192 MB L2 (pe rows are reused heavily).
// ---------------------------------------------------------------------------
__global__ __launch_bounds__(256)
void pe_add_kernel(const float* __restrict__ x,
                   const int* __restrict__ pos,
                   const float* __restrict__ pe,
                   float* __restrict__ out) {
  const int lane    = threadIdx.x & 63;   // 0..63 : which float4 of the 256-dim row
  const int tokBlk  = threadIdx.x >> 6;   // 0..3  : token within block
  const long token  = (long)blockIdx.x * 4 + tokBlk;   // < BATCH*SEQ

  const size_t rowOff = (size_t)token * DIM + (size_t)lane * 4;
  const int p = pos[token];

  // Streaming x: non-temporal load (th:NT) — single use, don't pollute caches.
  const v4f xv = __builtin_nontemporal_load(
      reinterpret_cast<const v4f*>(x + rowOff));

  // pe gather: regular temporal load — hot in L2 (4 MiB table, 192 MB L2).
  const v4f pv = *reinterpret_cast<const v4f*>(
      pe + (size_t)p * DIM + (size_t)lane * 4);

  v4f o;
  o.x = xv.x + pv.x;
  o.y = xv.y + pv.y;
  o.z = xv.z + pv.z;
  o.w = xv.w + pv.w;

  // Streaming store: non-temporal (th:NT), write-combining to HBM.
  __builtin_nontemporal_store(o, reinterpret_cast<v4f*>(out + rowOff));
}

// ---------------------------------------------------------------------------
// Harness entry point
//   d_in[0] = x    (float32, [32,4096,256])
//   d_in[1] = mask (bool,    [32,4096])  -- 1 byte per element
//   d_in[2] = pe   (float32, [4096,256])
//   d_out   = out  (float32, [32,4096,256])
//   d_ws    : >= 32*4096*4 bytes used for the pos[] table
// ---------------------------------------------------------------------------
extern "C" void kernel_launch(void* const* d_in, const int* in_sizes, int n_in,
                              void* d_out, int out_size, void* d_ws, size_t ws_size,
                              hipStream_t stream) {
  const float*         x    = (const float*)d_in[0];
  const unsigned char* mask = (const unsigned char*)d_in[1];
  const float*         pe   = (const float*)d_in[2];
  float*               out  = (float*)d_out;
  int*                  pos = (int*)d_ws;   // 32*4096*4 = 512 KiB scratch

  // Kernel A: 32 rows, 1024 threads each (one block per row).
  pe_pos_scan_kernel<<<BATCH, 1024, 0, stream>>>(mask, pos);

  // Kernel B: 4 tokens per 256-thread block.
  const int totalTokens = BATCH * SEQ;            // 131072
  const int blocks      = totalTokens / 4;        // 32768
  pe_add_kernel<<<blocks, 256, 0, stream>>>(x, pos, pe, out);
}